// IrradiationSingleTimestep_35364760715417
// MI455X (gfx1250) — compile-verified
//
#include <hip/hip_runtime.h>
#include <math.h>

// Irradiation phase-field single timestep, B x 1024 x 1024 periodic grid, fp32.
// Two-pass, bandwidth-bound stencil (~400 MB HBM traffic -> ~17us floor @23.3TB/s):
//   pass1: dF_dcv, dF_dci -> d_ws ; eta_new -> d_out[2*NE..)
//   pass2: lap(dF_*) + mobility -> cv_new, ci_new -> d_out[0..2*NE)
// LDS row staging via CDNA5 async global->LDS copies (ASYNCcnt-tracked
// global_load_async_to_lds_b128 + s_wait_asynccnt).

#define TILE_W 1024           // row length (W)
#define GRID_H 1024           // rows per batch image (H)
#define EPS_F  1e-6f
#define DT_F   1e-2f

#ifndef __has_builtin
#define __has_builtin(x) 0
#endif

#if __has_builtin(__builtin_amdgcn_global_load_async_to_lds_b128)
#define HAVE_ASYNC_LDS 1
#endif

// Pointee type per the compiler's builtin signature: v4i = int vector_size(16),
// in addrspace(1) (global) / addrspace(3) (LDS).
typedef int v4i __attribute__((vector_size(16)));
typedef __attribute__((address_space(1))) v4i as1_v4i;
typedef __attribute__((address_space(3))) v4i as3_v4i;

// Stage 16 contiguous bytes global -> LDS (per-lane async copy, ASYNCcnt).
__device__ __forceinline__ void stage16(const float* __restrict__ g, float* l) {
#if defined(HAVE_ASYNC_LDS)
  // Flat->AS1 is a bitcast; flat LDS pointer's low 32 bits are the AS3 offset
  // (addrspacecast AS3->flat = {aperture_hi, lds_offset}).
  __builtin_amdgcn_global_load_async_to_lds_b128(
      (as1_v4i*)(unsigned long long)(size_t)g,
      (as3_v4i*)(unsigned int)(size_t)l,
      0, 0);
#else
  *reinterpret_cast<float4*>(l) = *reinterpret_cast<const float4*>(g);
#endif
}

__device__ __forceinline__ void stage_wait() {
#if defined(HAVE_ASYNC_LDS)
#if __has_builtin(__builtin_amdgcn_s_wait_asynccnt)
  __builtin_amdgcn_s_wait_asynccnt(0);
#else
  asm volatile("s_wait_asynccnt 0" ::: "memory");
#endif
#endif
  __syncthreads();
}

__device__ __forceinline__ float clamp01(float v) {
  return fminf(fmaxf(v, 0.0f), 1.0f);
}

__device__ __forceinline__ float safelog(float m) {
  return __logf(m < EPS_F ? EPS_F : m);   // native v_log_f32 path
}

// ---------------------------------------------------------------- pass 1 ----
// grid = (H, B): y = blockIdx.x, b = blockIdx.y -> no integer division.
__global__ __launch_bounds__(256) void irr_pass1(
    const float* __restrict__ cv, const float* __restrict__ ci,
    const float* __restrict__ eta,
    const float* __restrict__ p_ev, const float* __restrict__ p_ei,
    const float* __restrict__ p_kbt, const float* __restrict__ p_kv,
    const float* __restrict__ p_ki, const float* __restrict__ p_keta,
    const float* __restrict__ p_L,
    float* __restrict__ dFv, float* __restrict__ dFi,
    float* __restrict__ eta_out)
{
  const int t  = threadIdx.x;            // 0..255
  const int y  = blockIdx.x;             // 0..H-1
  const int b  = blockIdx.y;             // 0..B-1
  const int ym = (y == 0) ? (GRID_H - 1) : (y - 1);
  const int yp = (y == GRID_H - 1) ? 0 : (y + 1);
  const size_t base  = (size_t)b << 20;            // b * H * W (1024*1024)
  const size_t rowY  = base + ((size_t)y  << 10);  // * TILE_W
  const size_t rowYm = base + ((size_t)ym << 10);
  const size_t rowYp = base + ((size_t)yp << 10);

  // 9 rows: cv(ym,y,yp), ci(ym,y,yp), eta(ym,y,yp) -> 36 KB LDS
  __shared__ float s[9][TILE_W];
  const int c4 = t * 4;                  // each thread stages one 16B chunk/row
  stage16(cv  + rowYm + c4, &s[0][c4]);
  stage16(cv  + rowY  + c4, &s[1][c4]);
  stage16(cv  + rowYp + c4, &s[2][c4]);
  stage16(ci  + rowYm + c4, &s[3][c4]);
  stage16(ci  + rowY  + c4, &s[4][c4]);
  stage16(ci  + rowYp + c4, &s[5][c4]);
  stage16(eta + rowYm + c4, &s[6][c4]);
  stage16(eta + rowY  + c4, &s[7][c4]);
  stage16(eta + rowYp + c4, &s[8][c4]);

  // uniform scalar params (s_load), overlapped with the async fills
  const float ev   = fabsf(p_ev[0])   + 0.001f;
  const float ei   = fabsf(p_ei[0])   + 0.001f;
  const float kbt  = fabsf(p_kbt[0])  + 0.001f;
  const float kv   = fabsf(p_kv[0])   + 0.001f;
  const float ki   = fabsf(p_ki[0])   + 0.001f;
  const float keta = fabsf(p_keta[0]) + 0.001f;
  const float L    = fabsf(p_L[0])    + 0.001f;

  stage_wait();

#pragma unroll
  for (int k = 0; k < 4; ++k) {
    const int x  = t + 256 * k;          // lane-per-bank: conflict-free LDS
    const int xm = (x - 1) & (TILE_W - 1);
    const int xp = (x + 1) & (TILE_W - 1);

    const float c_v = s[1][x], c_i = s[4][x], e = s[7][x];
    const float lap_v = s[1][xm] + s[1][xp] + s[0][x] + s[2][x] - 4.0f * c_v;
    const float lap_i = s[4][xm] + s[4][xp] + s[3][x] + s[5][x] - 4.0f * c_i;
    const float lap_e = s[7][xm] + s[7][xp] + s[6][x] + s[8][x] - 4.0f * e;

    const float em1 = e - 1.0f;
    const float h   = em1 * em1;
    const float jj  = e * e;
    const float cs  = 1.0f - c_v - c_i;
    const float lcv = safelog(c_v);
    const float lci = safelog(c_i);
    const float lcs = safelog(cs);

    const float fs = ev * c_v + ei * c_i +
                     kbt * (c_v * lcv + c_i * lci + cs * lcs);
    const float dfs_dcv = ev + kbt * (lcv - lcs);
    const float dfs_dci = ei + kbt * (lci - lcs);

    const float cvm1 = c_v - 1.0f;
    const float fv = cvm1 * cvm1 + c_i * c_i;

    const float dF_dcv = h * dfs_dcv + jj * (2.0f * cvm1) - kv * lap_v;
    const float dF_dci = h * dfs_dci + jj * (2.0f * c_i)  - ki * lap_i;
    const float dF_de  = fs * (2.0f * em1) + fv * (2.0f * e) - keta * lap_e;

    const size_t idx = rowY + x;         // coalesced 128B per wave
    dFv[idx] = dF_dcv;
    dFi[idx] = dF_dci;
    eta_out[idx] = clamp01(e - DT_F * L * dF_de);
  }
}

// ---------------------------------------------------------------- pass 2 ----
__global__ __launch_bounds__(256) void irr_pass2(
    const float* __restrict__ cv, const float* __restrict__ ci,
    const float* __restrict__ dFv, const float* __restrict__ dFi,
    const float* __restrict__ p_kbt, const float* __restrict__ p_dv,
    const float* __restrict__ p_di,
    float* __restrict__ cv_out, float* __restrict__ ci_out)
{
  const int t  = threadIdx.x;
  const int y  = blockIdx.x;
  const int b  = blockIdx.y;
  const int ym = (y == 0) ? (GRID_H - 1) : (y - 1);
  const int yp = (y == GRID_H - 1) ? 0 : (y + 1);
  const size_t base  = (size_t)b << 20;
  const size_t rowY  = base + ((size_t)y  << 10);
  const size_t rowYm = base + ((size_t)ym << 10);
  const size_t rowYp = base + ((size_t)yp << 10);

  // 6 rows: dFv(ym,y,yp), dFi(ym,y,yp) -> 24 KB LDS
  __shared__ float s[6][TILE_W];
  const int c4 = t * 4;
  stage16(dFv + rowYm + c4, &s[0][c4]);
  stage16(dFv + rowY  + c4, &s[1][c4]);
  stage16(dFv + rowYp + c4, &s[2][c4]);
  stage16(dFi + rowYm + c4, &s[3][c4]);
  stage16(dFi + rowY  + c4, &s[4][c4]);
  stage16(dFi + rowYp + c4, &s[5][c4]);

  const float kbt = fabsf(p_kbt[0]) + 0.001f;
  const float dv  = fabsf(p_dv[0])  + 0.001f;
  const float di  = fabsf(p_di[0])  + 0.001f;
  const float inv_kbt = 1.0f / kbt;     // uniform reciprocal, computed once

  stage_wait();

#pragma unroll
  for (int k = 0; k < 4; ++k) {
    const int x  = t + 256 * k;
    const int xm = (x - 1) & (TILE_W - 1);
    const int xp = (x + 1) & (TILE_W - 1);

    const float lap_v = s[1][xm] + s[1][xp] + s[0][x] + s[2][x] - 4.0f * s[1][x];
    const float lap_i = s[4][xm] + s[4][xp] + s[3][x] + s[5][x] - 4.0f * s[4][x];

    const size_t idx = rowY + x;
    const float c_v = cv[idx];          // pointwise only -> straight global load
    const float c_i = ci[idx];
    const float mv = dv * c_v * inv_kbt;
    const float mi = di * c_i * inv_kbt;

    cv_out[idx] = clamp01(c_v + DT_F * (mv * lap_v));
    ci_out[idx] = clamp01(c_i + DT_F * (mi * lap_i));
  }
}

// -------------------------------------------------------------- launcher ----
extern "C" void kernel_launch(void* const* d_in, const int* in_sizes, int n_in,
                              void* d_out, int out_size, void* d_ws, size_t ws_size,
                              hipStream_t stream) {
  const float* cv  = (const float*)d_in[0];
  const float* ci  = (const float*)d_in[1];
  const float* eta = (const float*)d_in[2];
  const float* p_ev   = (const float*)d_in[3];
  const float* p_ei   = (const float*)d_in[4];
  const float* p_kbt  = (const float*)d_in[5];
  const float* p_kv   = (const float*)d_in[6];
  const float* p_ki   = (const float*)d_in[7];
  const float* p_keta = (const float*)d_in[8];
  const float* p_dv   = (const float*)d_in[9];
  const float* p_di   = (const float*)d_in[10];
  const float* p_L    = (const float*)d_in[11];

  const size_t NE = (size_t)in_sizes[0];          // B*H*W
  const int B = (int)(NE / ((size_t)GRID_H * TILE_W));

  float* out     = (float*)d_out;                  // [cv_new | ci_new | eta_new]
  float* cv_out  = out;
  float* ci_out  = out + NE;
  float* eta_out = out + 2 * NE;

  float* dFv = (float*)d_ws;                       // 2*NE floats of scratch
  float* dFi = dFv + NE;

  const dim3 block(256);
  const dim3 grid(GRID_H, (unsigned)B);            // one block per (y, b) row

  irr_pass1<<<grid, block, 0, stream>>>(cv, ci, eta,
                                        p_ev, p_ei, p_kbt, p_kv, p_ki, p_keta, p_L,
                                        dFv, dFi, eta_out);
  irr_pass2<<<grid, block, 0, stream>>>(cv, ci, dFv, dFi,
                                        p_kbt, p_dv, p_di,
                                        cv_out, ci_out);
}